// DiagonalSSMLayer_62302795596644
// MI455X (gfx1250) — compile-verified
//
#include <hip/hip_runtime.h>

typedef float v2f __attribute__((ext_vector_type(2)));
typedef float v8f __attribute__((ext_vector_type(8)));

#define BATCH 8
#define SEQ 4096
#define DM 1024
#define NS 4
#define LN_EPS 1e-5f
#define CHUNK 128
#define NCHUNK (SEQ / CHUNK) /* 32 */

// ---------------------------------------------------------------------------
// Kernel 1: u[r,n] = sum_d x[r,d] * B_w[n,d] + B_b[n]
// One wave per 32 rows (two 16-row WMMA tiles sharing the B operand).
// v_wmma_f32_16x16x4_f32, K stepped by 4.
// A layout (16x4 f32): lanes 0-15 -> M=lane, K={k0,k0+1}; lanes 16-31 -> M=lane-16, K={k0+2,k0+3}
// B layout (4x16 f32): lanes 0-15 -> N=lane, K={k0,k0+1}; lanes 16-31 -> N=lane-16, K={k0+2,k0+3}
// C/D layout (16x16):  VGPR v, lanes 0-15 -> (M=v, N=lane); lanes 16-31 -> (M=v+8, N=lane-16)
// B load is branch-free: lanes with sub>=4 load a clamped (valid) row and
// multiply by 0, avoiding per-iteration exec-mask save/restore.
// ---------------------------------------------------------------------------
__global__ __launch_bounds__(32) void k1_proj_u(const float* __restrict__ x,
                                                const float* __restrict__ Bw,
                                                const float* __restrict__ Bb,
                                                float* __restrict__ u) {
  const int rowbase = blockIdx.x * 32;
  const int lane = threadIdx.x;
  const int half = lane >> 4;   // which K-pair this lane carries
  const int sub  = lane & 15;

  v8f c0 = {}, c1 = {};
  const float* xrow0 = x + (size_t)(rowbase + sub) * DM + 2 * half;
  const float* xrow1 = xrow0 + (size_t)16 * DM;

  const int   nclamp = (sub < NS) ? sub : (NS - 1);
  const float bmask  = (sub < NS) ? 1.0f : 0.0f;
  const float* bptr  = Bw + (size_t)nclamp * DM + 2 * half;  // B_w[n][k]

  for (int k0 = 0; k0 < DM; k0 += 4) {
    v2f b;
    b.x = bptr[k0]     * bmask;
    b.y = bptr[k0 + 1] * bmask;
    v2f a0; a0.x = xrow0[k0]; a0.y = xrow0[k0 + 1];
    v2f a1; a1.x = xrow1[k0]; a1.y = xrow1[k0 + 1];
    c0 = __builtin_amdgcn_wmma_f32_16x16x4_f32(false, a0, false, b, (short)0, c0,
                                               false, false);
    c1 = __builtin_amdgcn_wmma_f32_16x16x4_f32(false, a1, false, b, (short)0, c1,
                                               false, false);
  }

  if (sub < NS) {
    float bb = Bb[sub];
#pragma unroll
    for (int v = 0; v < 8; ++v) {
      int row = rowbase + v + 8 * half;
      u[(size_t)row * NS + sub]        = c0[v] + bb;
      u[(size_t)(row + 16) * NS + sub] = c1[v] + bb;
    }
  }
}

// ---------------------------------------------------------------------------
// Kernel 2: chunked parallel scan  h_t = lam*h_{t-1} + u_t  (h_{-1} = 0)
// One block per batch; 128 threads = (32 chunks x 4 states).
// Phase A: per-chunk partial (assumes h_in = 0). Phase B: 32-step prefix across
// chunks with lam^CHUNK. Phase C: replay chunk with correct start, emit hs.
// ---------------------------------------------------------------------------
__global__ __launch_bounds__(128) void k2_scan(const float* __restrict__ u,
                                               const float* __restrict__ log_lambda,
                                               float* __restrict__ hs) {
  const int b = blockIdx.x;
  const int tid = threadIdx.x;
  const int c = tid >> 2;   // chunk 0..31
  const int n = tid & 3;    // state 0..3

  __shared__ float P[NCHUNK][NS];
  __shared__ float H0[NCHUNK][NS];

  const float lam = 1.0f / (1.0f + __expf(-log_lambda[n]));
  const float* ub = u + (size_t)b * SEQ * NS;
  const int t0 = c * CHUNK;

  float p = 0.0f;
  for (int j = 0; j < CHUNK; ++j) p = lam * p + ub[(size_t)(t0 + j) * NS + n];
  P[c][n] = p;
  __syncthreads();

  if (tid < NS) {
    float lamL = lam;
#pragma unroll
    for (int i = 0; i < 7; ++i) lamL *= lamL;  // lam^128
    float h = 0.0f;
    for (int cc = 0; cc < NCHUNK; ++cc) {
      H0[cc][tid] = h;
      h = lamL * h + P[cc][tid];
    }
  }
  __syncthreads();

  float h = H0[c][n];
  float* hb = hs + (size_t)b * SEQ * NS;
  for (int j = 0; j < CHUNK; ++j) {
    h = lam * h + ub[(size_t)(t0 + j) * NS + n];
    hb[(size_t)(t0 + j) * NS + n] = h;
  }
}

// ---------------------------------------------------------------------------
// Kernel 3: y = hs*C_w^T + C_b ; z = x + y ; LayerNorm(z) -> out
// One wave per 32 seq rows (two 16-row tiles sharing C_w/C_b/ln tile loads).
// One WMMA per 16-wide d-tile per seq-tile yields a full 16x16 z tile.
// Two passes over d (stats, then emit); pass 2 re-reads x/C_w from WGP cache.
// ---------------------------------------------------------------------------
__global__ __launch_bounds__(32) void k3_out(const float* __restrict__ x,
                                             const float* __restrict__ hs,
                                             const float* __restrict__ Cw,
                                             const float* __restrict__ Cb,
                                             const float* __restrict__ lnw,
                                             const float* __restrict__ lnb,
                                             float* __restrict__ out) {
  const int rowbase = blockIdx.x * 32;
  const int lane = threadIdx.x;
  const int half = lane >> 4;
  const int sub  = lane & 15;

  v2f a0, a1;
  {
    const float* hp0 = hs + (size_t)(rowbase + sub) * NS + 2 * half;
    a0.x = hp0[0]; a0.y = hp0[1];
    const float* hp1 = hp0 + (size_t)16 * NS;
    a1.x = hp1[0]; a1.y = hp1[1];
  }

  const float* xt = x + (size_t)rowbase * DM;
  v8f s0 = {}, q0 = {}, s1 = {}, q1 = {};

  // Pass 1: accumulate per-row sum and sum of squares of z = x + y
  for (int d0 = 0; d0 < DM; d0 += 16) {
    const float* cp = Cw + (size_t)(d0 + sub) * NS + 2 * half;  // C_w[d][k]
    v2f bcol; bcol.x = cp[0]; bcol.y = cp[1];
    float cb = Cb[d0 + sub];
    v8f c0 = {cb, cb, cb, cb, cb, cb, cb, cb};
    v8f c1 = c0;
    c0 = __builtin_amdgcn_wmma_f32_16x16x4_f32(false, a0, false, bcol, (short)0,
                                               c0, false, false);
    c1 = __builtin_amdgcn_wmma_f32_16x16x4_f32(false, a1, false, bcol, (short)0,
                                               c1, false, false);
#pragma unroll
    for (int v = 0; v < 8; ++v) {
      int m = v + 8 * half;
      float z0 = xt[(size_t)m * DM + d0 + sub] + c0[v];
      float z1 = xt[(size_t)(m + 16) * DM + d0 + sub] + c1[v];
      s0[v] += z0; q0[v] += z0 * z0;
      s1[v] += z1; q1[v] += z1 * z1;
    }
  }

  // Cross-lane reduce within each 16-lane half (rows live on fixed halves)
  v8f mean0, rstd0, mean1, rstd1;
#pragma unroll
  for (int v = 0; v < 8; ++v) {
    float a = s0[v], b = q0[v], e = s1[v], f = q1[v];
#pragma unroll
    for (int msk = 1; msk <= 8; msk <<= 1) {
      a += __shfl_xor(a, msk, 32);
      b += __shfl_xor(b, msk, 32);
      e += __shfl_xor(e, msk, 32);
      f += __shfl_xor(f, msk, 32);
    }
    float mu0 = a * (1.0f / DM);
    float mu1 = e * (1.0f / DM);
    mean0[v] = mu0;
    mean1[v] = mu1;
    rstd0[v] = rsqrtf(b * (1.0f / DM) - mu0 * mu0 + LN_EPS);
    rstd1[v] = rsqrtf(f * (1.0f / DM) - mu1 * mu1 + LN_EPS);
  }

  // Pass 2: recompute z, normalize, write out
  for (int d0 = 0; d0 < DM; d0 += 16) {
    const float* cp = Cw + (size_t)(d0 + sub) * NS + 2 * half;
    v2f bcol; bcol.x = cp[0]; bcol.y = cp[1];
    float cb = Cb[d0 + sub];
    v8f c0 = {cb, cb, cb, cb, cb, cb, cb, cb};
    v8f c1 = c0;
    c0 = __builtin_amdgcn_wmma_f32_16x16x4_f32(false, a0, false, bcol, (short)0,
                                               c0, false, false);
    c1 = __builtin_amdgcn_wmma_f32_16x16x4_f32(false, a1, false, bcol, (short)0,
                                               c1, false, false);
    float lw = lnw[d0 + sub];
    float lb = lnb[d0 + sub];
#pragma unroll
    for (int v = 0; v < 8; ++v) {
      int m = v + 8 * half;
      float z0 = xt[(size_t)m * DM + d0 + sub] + c0[v];
      float z1 = xt[(size_t)(m + 16) * DM + d0 + sub] + c1[v];
      out[(size_t)(rowbase + m) * DM + d0 + sub] =
          (z0 - mean0[v]) * rstd0[v] * lw + lb;
      out[(size_t)(rowbase + m + 16) * DM + d0 + sub] =
          (z1 - mean1[v]) * rstd1[v] * lw + lb;
    }
  }
}

// ---------------------------------------------------------------------------
extern "C" void kernel_launch(void* const* d_in, const int* in_sizes, int n_in,
                              void* d_out, int out_size, void* d_ws, size_t ws_size,
                              hipStream_t stream) {
  const float* x  = (const float*)d_in[0];
  const float* ll = (const float*)d_in[1];
  const float* Bw = (const float*)d_in[2];
  const float* Bb = (const float*)d_in[3];
  const float* Cw = (const float*)d_in[4];
  const float* Cb = (const float*)d_in[5];
  const float* lw = (const float*)d_in[6];
  const float* lb = (const float*)d_in[7];
  float* out = (float*)d_out;

  float* u  = (float*)d_ws;                       // 8*4096*4 floats (512 KB)
  float* hs = u + (size_t)BATCH * SEQ * NS;       // 8*4096*4 floats (512 KB)

  k1_proj_u<<<BATCH * SEQ / 32, 32, 0, stream>>>(x, Bw, Bb, u);
  k2_scan<<<BATCH, 128, 0, stream>>>(u, ll, hs);
  k3_out<<<BATCH * SEQ / 32, 32, 0, stream>>>(x, hs, Cw, Cb, lw, lb, out);
}